// PositionalEncoderModule_82978768159582
// MI455X (gfx1250) — compile-verified
//
#include <hip/hip_runtime.h>
#include <hip/hip_bf16.h>

// PositionalEncoder: out[b,s,d] = x[b,s,d] + pe[s,d]
//   x: [8, 4096, 1024] f32  ->  pure streaming add, 256 MB HBM traffic,
//   roofline ~11 us @ 23.3 TB/s. No matmul -> WMMA inapplicable; CDNA5 path
//   used here is the async global->LDS mover (ASYNCcnt) for VGPR-free
//   pipelined streaming.
//
// One block per sequence row s; thread t owns dims [4t,4t+3] (16B, b128).
// PE is computed ONCE per thread with raw hw v_exp/v_sin/v_cos (Cody-Waite
// range reduction; exp2 arg is in [-26.6, 0] so the libm overflow/denorm
// guard is provably dead -> use __builtin_amdgcn_exp2f directly) and reused
// across all 8 batch rows while input streams through a double-buffered LDS
// pipeline (global_load_async_to_lds_b128, NT). Stores are NT b128.

typedef float v4f __attribute__((ext_vector_type(4)));

#define SEQ     4096
#define DMODEL  1024
#define BATCH   8
#define THREADS 256   // = DMODEL/4 lanes; 8 wave32s per block

__global__ __launch_bounds__(THREADS)
void pe_add_kernel(const float* __restrict__ in, float* __restrict__ out) {
    // Double-buffered staging: 2 x 256 x 16B = 8 KB LDS
    __shared__ v4f buf[2][THREADS];

    const int s = blockIdx.x;
    const int t = threadIdx.x;

    const size_t batch_stride = (size_t)SEQ * DMODEL;      // elements
    const float* base = in + (size_t)s * DMODEL + 4 * t;   // row s, lane's 16B

    // LDS byte addresses of this thread's two slots (generic ptr low 32 bits
    // == workgroup-relative LDS byte offset per the flat aperture mapping).
    unsigned lds0 = (unsigned)(uintptr_t)&buf[0][t];
    unsigned lds1 = (unsigned)(uintptr_t)&buf[1][t];

    // Kick off async fills for batch rows 0 and 1 (ASYNCcnt = 2 outstanding).
    {
        unsigned long long a0 = (unsigned long long)(uintptr_t)(base);
        unsigned long long a1 = (unsigned long long)(uintptr_t)(base + batch_stride);
        asm volatile("global_load_async_to_lds_b128 %0, %1, off th:TH_LOAD_NT"
                     :: "v"(lds0), "v"(a0) : "memory");
        asm volatile("global_load_async_to_lds_b128 %0, %1, off th:TH_LOAD_NT"
                     :: "v"(lds1), "v"(a1) : "memory");
    }

    // ---- PE for this lane's 4 dims, overlapped with the async fills ----
    // inv_freq(d) = 10000^(-2d/D) = exp2(d * (-2*log2(10000)/D)), arg in
    // [-26.6, 0] -> raw v_exp_f32 is safe (no overflow/denorm fixup needed).
    // angle = s * inv_freq  (<= 4095 rad).  Cody-Waite reduction:
    //   n = rint(angle/2pi)  (n <= 652 < 2^10)
    //   2pi = 6.28125 (8-bit mantissa -> n*hi exact) + 1.9353071795e-3
    //   r = angle - n*hi - n*lo,  |r| <= pi
    // then native hw sin/cos on normalized turns r/2pi (|.| <= 0.5).
    const float kExpScale = -2.0f * 13.287712379549449f / (float)DMODEL;
    const float kInv2Pi   = 0.15915494309189535f;
    const float kTwoPiHi  = 6.28125f;
    const float kTwoPiLo  = 1.9353071795864769e-3f;
    const float fs = (float)s;
    v4f pev;
#pragma unroll
    for (int j = 0; j < 4; ++j) {
        int d = 4 * t + j;                         // d even for j even (4t even)
        float angle = fs * __builtin_amdgcn_exp2f((float)d * kExpScale);
        float n = __builtin_rintf(angle * kInv2Pi);
        float r = __builtin_fmaf(n, -kTwoPiHi, angle);
        r       = __builtin_fmaf(n, -kTwoPiLo, r);
        float turns = r * kInv2Pi;                 // |turns| <= 0.5
        pev[j] = (j & 1) ? __builtin_amdgcn_cosf(turns)
                         : __builtin_amdgcn_sinf(turns);
    }

    // ---- Pipelined consume: wait oldest fill, add PE, NT-store, refill ----
#pragma unroll
    for (int b = 0; b < BATCH; ++b) {
        if (b == BATCH - 1) {
            asm volatile("s_wait_asynccnt 0" ::: "memory");
        } else {
            asm volatile("s_wait_asynccnt 1" ::: "memory");
        }

        v4f x = buf[b & 1][t];                     // ds_load_b128 (own wave's slot)
        v4f r = x + pev;

        v4f* dst = (v4f*)(out + (size_t)b * batch_stride + (size_t)s * DMODEL + 4 * t);
        __builtin_nontemporal_store(r, dst);       // global_store_b128, NT hint

        if (b + 2 < BATCH) {                       // refill the buffer just drained
            unsigned long long an =
                (unsigned long long)(uintptr_t)(base + (size_t)(b + 2) * batch_stride);
            unsigned ldsn = (b & 1) ? lds1 : lds0;
            asm volatile("global_load_async_to_lds_b128 %0, %1, off th:TH_LOAD_NT"
                         :: "v"(ldsn), "v"(an) : "memory");
        }
    }
}

extern "C" void kernel_launch(void* const* d_in, const int* in_sizes, int n_in,
                              void* d_out, int out_size, void* d_ws, size_t ws_size,
                              hipStream_t stream) {
    const float* x = (const float*)d_in[0];
    float* out = (float*)d_out;
    (void)in_sizes; (void)n_in; (void)out_size; (void)d_ws; (void)ws_size;
    pe_add_kernel<<<dim3(SEQ), dim3(THREADS), 0, stream>>>(x, out);
}